// Linear_4698694222141
// MI455X (gfx1250) — compile-verified
//
#include <hip/hip_runtime.h>
#include <hip/hip_bf16.h>

#define UNITS 16
#define DCOLS 272                 // 16 + 16*16
#define ROWS_PER_WAVE 16
#define WAVES_PER_BLOCK 8
#define BLOCK_THREADS (WAVES_PER_BLOCK * 32)   // 256 threads = 8 wave32
#define ROWS_PER_BLOCK (WAVES_PER_BLOCK * ROWS_PER_WAVE)  // 128 rows

typedef __attribute__((ext_vector_type(2))) float v2f;
typedef __attribute__((ext_vector_type(8))) float v8f;

// Only check for the device builtin in the device compilation pass; the host
// pass has no AMDGCN builtins and would trip this spuriously.
#if defined(__HIP_DEVICE_COMPILE__) && defined(__AMDGCN__)
#if !__has_builtin(__builtin_amdgcn_wmma_f32_16x16x4_f32)
#error "missing __builtin_amdgcn_wmma_f32_16x16x4_f32 on this toolchain"
#endif
#endif

// out[m] = sum_k in[m,k] * w[k] + bias, computed as 16x16x4 f32 WMMA tiles.
// A (16x4): lanes 0-15 -> K=0,1 ; lanes 16-31 -> K=2,3 ; row M = lane%16.
// B (4x16): weight w[k] broadcast across all N columns (same per-lane-half K split).
// C/D (16x16 f32): VGPR i holds M=i (lanes 0-15) and M=i+8 (lanes 16-31).
__global__ __launch_bounds__(BLOCK_THREADS) void linear_pairwise_wmma(
    const float* __restrict__ in, const float* __restrict__ kern,
    const float* __restrict__ w1p, const float* __restrict__ w2p,
    const float* __restrict__ biasp, float* __restrict__ out, int n) {
  __shared__ float sw[DCOLS];

  // Recompute the 272-entry weight vector per block (cheap, avoids a 2nd kernel).
  const float w1 = w1p[0];
  const float w2 = w2p[0];
  for (int j = threadIdx.x; j < DCOLS; j += BLOCK_THREADS) {
    float v;
    if (j < UNITS) {
      v = kern[j];
    } else {
      const int p  = j - UNITS;
      const int x1 = p >> 4, x2 = p & 15;
      const int a1 = x1 >> 2, b1 = x1 & 3;
      const int a2 = x2 >> 2, b2 = x2 & 3;
      v = kern[4 * a1 + a2] * kern[4 * b1 + b2] * w1 +
          kern[4 * a1 + b2] * kern[4 * b1 + a2] * w2;
    }
    sw[j] = v;
  }
  __syncthreads();

  const int lane     = threadIdx.x & 31;
  const int wave     = threadIdx.x >> 5;
  const int tileBase = blockIdx.x * ROWS_PER_BLOCK + wave * ROWS_PER_WAVE;

  int m = tileBase + (lane & 15);
  if (m > n - 1) m = n - 1;          // clamp tail rows (stores are guarded below)
  const int koff = (lane >> 4) << 1; // 0 for lanes 0-15, 2 for lanes 16-31

  const float* __restrict__ rowp = in + (size_t)m * DCOLS + koff;
  const float* wp = sw + koff;

  v8f c = {0.f, 0.f, 0.f, 0.f, 0.f, 0.f, 0.f, 0.f};

#pragma unroll 4
  for (int k0 = 0; k0 < DCOLS; k0 += 4) {
    const float2 av = *(const float2*)(rowp + k0);  // global_load_b64, 8B aligned
    const float2 bv = *(const float2*)(wp + k0);    // ds_load_b64, 8B aligned
    v2f a; a[0] = av.x; a[1] = av.y;
    v2f b; b[0] = bv.x; b[1] = bv.y;
    c = __builtin_amdgcn_wmma_f32_16x16x4_f32(
        /*neg_a=*/false, a, /*neg_b=*/false, b,
        /*c_mod=*/(short)0, c, /*reuse_a=*/false, /*reuse_b=*/false);
  }

  const float bias = biasp[0];
  // Every D column holds the same dot product; lane 0 writes M=0..7,
  // lane 16 writes M=8..15 of this tile.
  if ((lane & 15) == 0) {
    const int mbase = tileBase + ((lane >> 4) << 3);
#pragma unroll
    for (int i = 0; i < 8; ++i) {
      const int mm = mbase + i;
      if (mm < n) out[mm] = c[i] + bias;
    }
  }
}

extern "C" void kernel_launch(void* const* d_in, const int* in_sizes, int n_in,
                              void* d_out, int out_size, void* d_ws, size_t ws_size,
                              hipStream_t stream) {
  const float* in   = (const float*)d_in[0];   // [N, 272]
  const float* kern = (const float*)d_in[1];   // [16]
  const float* w1   = (const float*)d_in[2];   // [1]
  const float* w2   = (const float*)d_in[3];   // [1]
  const float* bias = (const float*)d_in[4];   // [1]
  float* out = (float*)d_out;                  // [N]

  const int n = in_sizes[0] / DCOLS;
  const int blocks = (n + ROWS_PER_BLOCK - 1) / ROWS_PER_BLOCK;
  linear_pairwise_wmma<<<blocks, BLOCK_THREADS, 0, stream>>>(in, kern, w1, w2,
                                                             bias, out, n);
}